// GatedGCNBlock_3959959847627
// MI455X (gfx1250) — compile-verified
//
#include <hip/hip_runtime.h>
#include <math.h>

#define N_NODES    50000
#define DIM        128
#define NUM_LAYERS 3
#define LDS_PITCH  132   // 128 + 4 skew: kills LDS bank conflicts on row-strided b64 reads

typedef __attribute__((ext_vector_type(2))) float v2f;
typedef __attribute__((ext_vector_type(8))) float v8f;

// ---------------------------------------------------------------------------
// WMMA f32 GEMM:  C[M, Nc] = A[M, 128] * B (+ bias[n])
//   BT = false : B stored row-major [128, Nc]      (h @ W_i)
//   BT = true  : B stored as [Nc, 128], use B^T    (x @ W^T, torch-style weights)
// One wave -> one 16x16 tile via 32x v_wmma_f32_16x16x4_f32 (K = 128).
// 8 waves / block share one 16x128 A tile staged in LDS.
// Per CDNA5 ISA layouts (CDNA5 ISA):
//   A: lane l holds A[l%16][kb + 2*(l>>4) + {0,1}]  (2 VGPRs = float2)
//   B: lane l holds B[kb + 2*(l>>4) + {0,1}][l%16]
//   C/D: VGPR v -> row v + 8*(l>>4), col l%16
// ---------------------------------------------------------------------------
template <bool BT>
__global__ __launch_bounds__(256) void gemm_wmma_f32(
    const float* __restrict__ A, const float* __restrict__ B,
    const float* __restrict__ bias, float* __restrict__ C, int Nc)
{
    __shared__ float As[16 * LDS_PITCH];

    const int tid    = threadIdx.x;
    const int tile_m = blockIdx.x;

    // Stage the 16x128 A tile (coalesced: 2 rows per 256-thread sweep).
    {
        const float* Ab = A + (size_t)tile_m * 16 * DIM;
        #pragma unroll
        for (int i = tid; i < 16 * DIM; i += 256)
            As[(i >> 7) * LDS_PITCH + (i & 127)] = Ab[i];
    }
    __syncthreads();

    const int wave  = tid >> 5;
    const int lane  = tid & 31;
    const int l16   = lane & 15;
    const int khalf = (lane >> 4) << 1;          // 0 or 2
    const int n     = blockIdx.y * 128 + wave * 16 + l16;

    v8f acc = {};
    #pragma unroll
    for (int kb = 0; kb < DIM; kb += 4) {
        const int k0 = kb + khalf;
        // A fragment from LDS (b64 load, conflict-free thanks to pitch skew)
        v2f a;
        {
            const float2 t = *(const float2*)&As[l16 * LDS_PITCH + k0];
            a.x = t.x; a.y = t.y;
        }
        // B fragment from global (weights are tiny -> resident in WGP$/L2)
        v2f b;
        if (BT) {
            const float2 t = *(const float2*)(B + (size_t)n * DIM + k0);
            b.x = t.x; b.y = t.y;
        } else {
            b.x = B[(size_t)k0 * Nc + n];
            b.y = B[(size_t)(k0 + 1) * Nc + n];
        }
        acc = __builtin_amdgcn_wmma_f32_16x16x4_f32(
            /*neg_a=*/false, a, /*neg_b=*/false, b,
            /*c_mod=*/(short)0, acc, /*reuse_a=*/false, /*reuse_b=*/false);
    }

    const float bv   = bias ? bias[n] : 0.0f;
    const int   row0 = (lane >> 4) << 3;         // 0 or 8
    const size_t m0  = (size_t)tile_m * 16 + row0;
    #pragma unroll
    for (int v = 0; v < 8; ++v)
        C[(m0 + v) * (size_t)Nc + n] = acc[v] + bv;
}

// ---------------------------------------------------------------------------
// agg = 0
// ---------------------------------------------------------------------------
__global__ void zero_f4(float4* __restrict__ p, int n4)
{
    int i = blockIdx.x * blockDim.x + threadIdx.x;
    if (i < n4) p[i] = make_float4(0.f, 0.f, 0.f, 0.f);
}

// ---------------------------------------------------------------------------
// One wave per edge: agg[dst[e]] += m[src[e]]  (128 floats = 32 lanes x float4)
// float4 gather + 4x global_atomic_add_f32 per lane; hits L2 (agg is 25.6 MB).
// ---------------------------------------------------------------------------
__global__ __launch_bounds__(256) void scatter_add_edges(
    const float* __restrict__ m, const int* __restrict__ src,
    const int* __restrict__ dst, float* __restrict__ agg, int E)
{
    const int e = (blockIdx.x * blockDim.x + threadIdx.x) >> 5;
    if (e >= E) return;
    const int lane = threadIdx.x & 31;
    const int s = src[e];
    const int d = dst[e];
    const float4 v = ((const float4*)(m + (size_t)s * DIM))[lane];
    float* a = agg + (size_t)d * DIM + lane * 4;
    atomicAdd(a + 0, v.x);
    atomicAdd(a + 1, v.y);
    atomicAdd(a + 2, v.z);
    atomicAdd(a + 3, v.w);
}

// ---------------------------------------------------------------------------
// GRU cell (torch gate order r,z,n), in-place update of h (per-element safe).
// ---------------------------------------------------------------------------
__global__ void gru_kernel(const float* __restrict__ gi, const float* __restrict__ gh,
                           float* __restrict__ h, int total)
{
    int i = blockIdx.x * blockDim.x + threadIdx.x;
    if (i >= total) return;
    const int node = i >> 7;          // /DIM
    const int d    = i & (DIM - 1);
    const size_t b = (size_t)node * 3 * DIM + d;
    const float gir = gi[b], giz = gi[b + DIM], gin = gi[b + 2 * DIM];
    const float ghr = gh[b], ghz = gh[b + DIM], ghn = gh[b + 2 * DIM];
    const float r  = 1.0f / (1.0f + __expf(-(gir + ghr)));
    const float z  = 1.0f / (1.0f + __expf(-(giz + ghz)));
    const float nn = tanhf(gin + r * ghn);
    h[i] = (1.0f - z) * nn + z * h[i];
}

__global__ void relu_kernel(const float* __restrict__ h, float* __restrict__ out, int total)
{
    int i = blockIdx.x * blockDim.x + threadIdx.x;
    if (i < total) out[i] = fmaxf(h[i], 0.0f);
}

// ---------------------------------------------------------------------------
// Launch
// ---------------------------------------------------------------------------
extern "C" void kernel_launch(void* const* d_in, const int* in_sizes, int n_in,
                              void* d_out, int out_size, void* d_ws, size_t ws_size,
                              hipStream_t stream)
{
    const float* x      = (const float*)d_in[0];
    const int*   ei     = (const int*)  d_in[1];   // [2, E] flat
    const float* lin_w  = (const float*)d_in[2];   // [D, D]
    const float* lin_b  = (const float*)d_in[3];   // [D]
    const float* ggnn_w = (const float*)d_in[4];   // [L, D, D]
    const float* w_ih   = (const float*)d_in[5];   // [3D, D]
    const float* w_hh   = (const float*)d_in[6];   // [3D, D]
    const float* b_ih   = (const float*)d_in[7];   // [3D]
    const float* b_hh   = (const float*)d_in[8];   // [3D]

    const int E   = in_sizes[1] / 2;
    const int* src = ei;
    const int* dst = ei + E;

    // Carve workspace (256B aligned): h, m, agg [N*D], gi, gh [N*3D]
    char* w = (char*)d_ws;
    auto carve = [&](size_t bytes) -> float* {
        char* p = w;
        w += (bytes + 255) & ~(size_t)255;
        return (float*)p;
    };
    float* h   = carve((size_t)N_NODES * DIM * 4);
    float* mb  = carve((size_t)N_NODES * DIM * 4);
    float* agg = carve((size_t)N_NODES * DIM * 4);
    float* gi  = carve((size_t)N_NODES * 3 * DIM * 4);
    float* gh  = carve((size_t)N_NODES * 3 * DIM * 4);

    const dim3 blk(256);
    const dim3 g128(N_NODES / 16, 1);       // 3125 M-tiles, 128 cols
    const dim3 g384(N_NODES / 16, 3);       // 3125 M-tiles, 384 cols
    const int  total = N_NODES * DIM;
    const int  nd4   = total / 4;
    const int  eblk  = (int)(((long long)E * 32 + 255) / 256);

    // h = x @ lin_w^T + lin_b
    gemm_wmma_f32<true><<<g128, blk, 0, stream>>>(x, lin_w, lin_b, h, DIM);

    for (int layer = 0; layer < NUM_LAYERS; ++layer) {
        // m = h @ W_layer (row-major B)
        gemm_wmma_f32<false><<<g128, blk, 0, stream>>>(
            h, ggnn_w + (size_t)layer * DIM * DIM, nullptr, mb, DIM);
        // agg = segment_sum(m[src], dst)
        zero_f4<<<(nd4 + 255) / 256, blk, 0, stream>>>((float4*)agg, nd4);
        scatter_add_edges<<<eblk, blk, 0, stream>>>(mb, src, dst, agg, E);
        // gi = agg @ w_ih^T + b_ih ; gh = h @ w_hh^T + b_hh
        gemm_wmma_f32<true><<<g384, blk, 0, stream>>>(agg, w_ih, b_ih, gi, 3 * DIM);
        gemm_wmma_f32<true><<<g384, blk, 0, stream>>>(h,   w_hh, b_hh, gh, 3 * DIM);
        // h = GRU(agg-gates, h-gates, h)
        gru_kernel<<<(total + 255) / 256, blk, 0, stream>>>(gi, gh, h, total);
    }

    relu_kernel<<<(total + 255) / 256, blk, 0, stream>>>(h, (float*)d_out, total);
}